// GCNConv_41790031790243
// MI455X (gfx1250) — compile-verified
//
#include <hip/hip_runtime.h>

typedef __attribute__((ext_vector_type(2))) float v2f;
typedef __attribute__((ext_vector_type(8))) float v8f;

#define K_DIM 256          // input features
#define N_OUT 128          // output features
#define LDS_STRIDE 260     // 256 + 4 pad -> conflict-free (260 mod 64 = 4)

// -------------------------------------------------------------------------
// Phase 1: h = x @ W + b   (fp32 WMMA 16x16x4)
// Block = 256 threads = 8 waves. Block computes rows [16*bid, 16*bid+16) x all
// 128 output cols; wave w owns cols [16w, 16w+16). The 16x256 x-tile is staged
// in LDS once and shared by all 8 waves (x read from HBM exactly once).
// -------------------------------------------------------------------------
__global__ void __launch_bounds__(256) gcn_gemm_wmma(
    const float* __restrict__ x, const float* __restrict__ W,
    const float* __restrict__ bias, float* __restrict__ h, int n_nodes) {
  __shared__ float xs[16 * LDS_STRIDE];

  const int tid  = threadIdx.x;
  const int row0 = blockIdx.x * 16;

  // Cooperative load of the 16x256 A tile: 16 threads per row, 4 float4 each.
  {
    const int r   = tid >> 4;          // 0..15 row within tile
    const int seg = tid & 15;          // 0..15 segment along K
    int srow = row0 + r;
    if (srow >= n_nodes) srow = n_nodes - 1;   // clamp (N divisible by 16 anyway)
    const float4* src = (const float4*)(x + (size_t)srow * K_DIM);
    float4* dst = (float4*)(xs + r * LDS_STRIDE);
#pragma unroll
    for (int j = 0; j < 4; ++j) {
      dst[seg * 4 + j] = src[seg * 4 + j];
    }
  }
  __syncthreads();

  const int wave = tid >> 5;
  const int lane = tid & 31;
  const int half = lane >> 4;          // lanes 16-31 carry K+2 (A/B), M+8 (C/D)
  const int lid  = lane & 15;
  const int ncol = wave * 16 + lid;    // output column for B/C/D fragments

  v8f acc = {};
#pragma unroll 8
  for (int k0 = 0; k0 < K_DIM; k0 += 4) {
    const int kk = k0 + half * 2;
    v2f a, b;
    // A 16x4 f32 layout: lanes 0-15 -> K=0,1 ; lanes 16-31 -> K=2,3 ; M = lid
    a.x = xs[lid * LDS_STRIDE + kk];
    a.y = xs[lid * LDS_STRIDE + kk + 1];
    // B 4x16 f32 layout: row striped across lanes; VGPR0=K{0|2}, VGPR1=K{1|3}
    b.x = W[(size_t)kk * N_OUT + ncol];
    b.y = W[(size_t)(kk + 1) * N_OUT + ncol];
    acc = __builtin_amdgcn_wmma_f32_16x16x4_f32(
        /*neg_a=*/false, a, /*neg_b=*/false, b,
        /*c_mod=*/(short)0, acc, /*reuse_a=*/false, /*reuse_b=*/false);
  }

  const float bv = bias[ncol];
  // C/D layout: VGPR r -> M=r (lanes 0-15), M=r+8 (lanes 16-31); N = lid
  float* hp = h + (size_t)(row0 + half * 8) * N_OUT + ncol;
  if (row0 + 16 <= n_nodes) {
    // Fast path: whole tile in range -> 8 unguarded coalesced b32 stores.
#pragma unroll
    for (int r = 0; r < 8; ++r) {
      hp[(size_t)r * N_OUT] = acc[r] + bv;
    }
  } else {
#pragma unroll
    for (int r = 0; r < 8; ++r) {
      const int m = row0 + r + half * 8;
      if (m < n_nodes) h[(size_t)m * N_OUT + ncol] = acc[r] + bv;
    }
  }
}

// -------------------------------------------------------------------------
// Phase 2: out[dst] += w_e * h[src].  One wave per edge: lane holds 4 of the
// 128 features (float4 gather = global_load_b128), then 4 no-return
// global_atomic_add_f32 per lane. h/out are L2-resident (51.2 MB each).
// -------------------------------------------------------------------------
__global__ void __launch_bounds__(256) gcn_spmm_scatter(
    const long long* __restrict__ edge_index,   // [2, E]: row0=dst, row1=src
    const float* __restrict__ edge_weight,
    const float* __restrict__ h, float* __restrict__ out, int num_edges) {
  const int gwave = (int)((blockIdx.x * (size_t)blockDim.x + threadIdx.x) >> 5);
  const int lane  = threadIdx.x & 31;
  if (gwave >= num_edges) return;

  const long long dst = edge_index[gwave];
  const long long src = edge_index[(size_t)num_edges + gwave];
  const float w = edge_weight[gwave];

  const float4 v = ((const float4*)(h + (size_t)src * N_OUT))[lane];
  float* o = out + (size_t)dst * N_OUT + lane * 4;
  atomicAdd(o + 0, v.x * w);
  atomicAdd(o + 1, v.y * w);
  atomicAdd(o + 2, v.z * w);
  atomicAdd(o + 3, v.w * w);
}

extern "C" void kernel_launch(void* const* d_in, const int* in_sizes, int n_in,
                              void* d_out, int out_size, void* d_ws, size_t ws_size,
                              hipStream_t stream) {
  const float*     x  = (const float*)d_in[0];      // [N,256] fp32
  const long long* ei = (const long long*)d_in[1];  // [2,E] int64
  const float*     ew = (const float*)d_in[2];      // [E] fp32
  const float*     W  = (const float*)d_in[3];      // [256,128] fp32
  const float*     b  = (const float*)d_in[4];      // [128] fp32

  const int n_nodes = in_sizes[0] / K_DIM;
  const int E       = in_sizes[2];
  float* h   = (float*)d_ws;                        // [N,128] intermediate
  float* out = (float*)d_out;

  // Zero accumulation target (harness poisons d_out; deterministic each call).
  hipMemsetAsync(d_out, 0, (size_t)out_size * sizeof(float), stream);

  const int gemm_blocks = (n_nodes + 15) / 16;
  gcn_gemm_wmma<<<gemm_blocks, 256, 0, stream>>>(x, W, b, h, n_nodes);

  const int spmm_blocks = (E + 7) / 8;              // 8 waves (edges) per block
  gcn_spmm_scatter<<<spmm_blocks, 256, 0, stream>>>(ei, ew, h, out, E);
}